// BatchNorm2d_fi_30889404793382
// MI455X (gfx1250) — compile-verified
//
#include <hip/hip_runtime.h>
#include <hip/hip_bf16.h>
#include <stdint.h>

// BatchNorm2d + bitflip fault injection, MI455X (gfx1250, wave32).
// Pass 1: per-channel sum/sumsq via V_WMMA_F32_16X16X4_F32 (B = ones -> row-sum
//         accumulation on the XDL path), contiguous b128 streaming loads with
//         immediate offsets, 4 independent accumulators, WGP-scope prefetch.
// Pass 2: tiny per-channel stats fold (deterministic, no float atomics).
// Pass 3: streaming normalize + FI, non-temporal load/store (data set > 192MB L2).

typedef __attribute__((ext_vector_type(2))) float v2f;
typedef __attribute__((ext_vector_type(4))) float v4f;
typedef __attribute__((ext_vector_type(8))) float v8f;

#define C_CH   128
#define B_N    32
#define PLANE  12544                       // 112*112
#define SPLIT  8                           // reduce blocks per channel
#define PLANES_PER_BLOCK (B_N / SPLIT)     // 4 planes per block = 8 half-planes = 8 waves
#define HALF_TILES 49                      // tiles of 128 floats per half-plane (98/2)
#define N_PER_CH  ((float)((size_t)B_N * PLANE))               // 401408
#define TOTAL_ELEMS ((size_t)B_N * C_CH * PLANE)               // 51,380,224
#define EPSI 1e-5f
#define FI_THRESH 4294967u                 // floor(0.001 * 2^32)

// ws layout (floats):
//   [0,    1024) : per-(channel,split) partial sums
//   [1024, 2048) : per-(channel,split) partial sumsq
//   [2048, 2176) : per-channel mean
//   [2176, 2304) : per-channel invstd

__device__ __forceinline__ uint32_t fi_hash(uint32_t x) {
    x ^= x >> 16; x *= 0x7feb352du;
    x ^= x >> 15; x *= 0x846ca68bu;
    x ^= x >> 16;
    return x;
}

__device__ __forceinline__ float fi(float v, uint32_t idx, uint32_t stage) {
    uint32_t h = fi_hash(idx * 0x9E3779B1u + stage * 0x85EBCA77u);
    return (h < FI_THRESH) ? __uint_as_float(__float_as_uint(v) ^ 0x40000000u) : v;
}

// Full-wave (32 lane) allreduce-add via ds_swizzle butterflies.
// offset[15]=0 group-of-32 mode: and_mask=0x1f, or_mask=0, xor_mask=offset[14:10].
__device__ __forceinline__ float wave_reduce_add(float v) {
    v += __uint_as_float(__builtin_amdgcn_ds_swizzle(__float_as_uint(v), 0x401f)); // xor 16
    v += __uint_as_float(__builtin_amdgcn_ds_swizzle(__float_as_uint(v), 0x201f)); // xor 8
    v += __uint_as_float(__builtin_amdgcn_ds_swizzle(__float_as_uint(v), 0x101f)); // xor 4
    v += __uint_as_float(__builtin_amdgcn_ds_swizzle(__float_as_uint(v), 0x081f)); // xor 2
    v += __uint_as_float(__builtin_amdgcn_ds_swizzle(__float_as_uint(v), 0x041f)); // xor 1
    return v;
}

__global__ __launch_bounds__(256)
void bn_reduce_kernel(const float* __restrict__ x, float* __restrict__ ws) {
    const int c    = blockIdx.x;           // channel
    const int s    = blockIdx.y;           // split
    const int lane = threadIdx.x & 31;
    const int wave = threadIdx.x >> 5;

    // Wave w owns half-plane: plane = s*4 + (w>>1), half = w&1. Fully contiguous
    // 49 tiles * 512B -> pure base + immediate-offset b128 loads in the loop.
    const int p = s * PLANES_PER_BLOCK + (wave >> 1);
    const float* base = x + ((size_t)p * C_CH + c) * PLANE
                          + (wave & 1) * (HALF_TILES * 128) + lane * 4;

    const v2f ones = {1.0f, 1.0f};
    v8f accS0 = {0.f,0.f,0.f,0.f,0.f,0.f,0.f,0.f};
    v8f accS1 = {0.f,0.f,0.f,0.f,0.f,0.f,0.f,0.f};
    v8f accQ0 = {0.f,0.f,0.f,0.f,0.f,0.f,0.f,0.f};
    v8f accQ1 = {0.f,0.f,0.f,0.f,0.f,0.f,0.f,0.f};

    #pragma unroll 7
    for (int it = 0; it < HALF_TILES; ++it) {
        v4f v = *reinterpret_cast<const v4f*>(base + it * 128);  // global_load_b128
        __builtin_prefetch(base + (it + 8) * 128, 0, 3);         // WGP-scope prefetch
        v2f a0 = { v.x, v.y };
        v2f a1 = { v.z, v.w };
        v2f q0 = a0 * a0;
        v2f q1 = a1 * a1;
        // D[m][n] += sum_k A[m][k]  (B = ones). 4 independent C-chains.
        accS0 = __builtin_amdgcn_wmma_f32_16x16x4_f32(false, a0, false, ones, (short)0, accS0, false, false);
        accS1 = __builtin_amdgcn_wmma_f32_16x16x4_f32(false, a1, false, ones, (short)0, accS1, false, false);
        accQ0 = __builtin_amdgcn_wmma_f32_16x16x4_f32(false, q0, false, ones, (short)0, accQ0, false, false);
        accQ1 = __builtin_amdgcn_wmma_f32_16x16x4_f32(false, q1, false, ones, (short)0, accQ1, false, false);
    }

    v8f accS = accS0 + accS1;
    v8f accQ = accQ0 + accQ1;

    // Each row-sum is replicated across 16 D columns -> total = sum(all entries)/16.
    float sSum = accS[0]+accS[1]+accS[2]+accS[3]+accS[4]+accS[5]+accS[6]+accS[7];
    float qSum = accQ[0]+accQ[1]+accQ[2]+accQ[3]+accQ[4]+accQ[5]+accQ[6]+accQ[7];
    sSum = wave_reduce_add(sSum);
    qSum = wave_reduce_add(qSum);

    __shared__ float shS[8], shQ[8];
    if (lane == 0) { shS[wave] = sSum; shQ[wave] = qSum; }
    __syncthreads();
    if (threadIdx.x == 0) {
        float ts = 0.f, tq = 0.f;
        #pragma unroll
        for (int k = 0; k < 8; ++k) { ts += shS[k]; tq += shQ[k]; }
        ws[c * SPLIT + s]                = ts * 0.0625f;   // /16 replication factor
        ws[C_CH * SPLIT + c * SPLIT + s] = tq * 0.0625f;
    }
}

__global__ __launch_bounds__(128)
void bn_stats_kernel(float* __restrict__ ws) {
    const int c = threadIdx.x;             // 128 threads, one per channel
    float sum = 0.f, sq = 0.f;
    #pragma unroll
    for (int s = 0; s < SPLIT; ++s) {
        sum += ws[c * SPLIT + s];
        sq  += ws[C_CH * SPLIT + c * SPLIT + s];
    }
    const float mean = sum / N_PER_CH;
    const float var  = sq / N_PER_CH - mean * mean;   // biased variance
    ws[2048 + c]        = mean;
    ws[2048 + C_CH + c] = rsqrtf(var + EPSI);
}

__global__ __launch_bounds__(256)
void bn_norm_kernel(const float* __restrict__ x, const float* __restrict__ wgt,
                    const float* __restrict__ bias, const float* __restrict__ ws,
                    float* __restrict__ out) {
    const size_t i4 = ((size_t)blockIdx.x * 256 + threadIdx.x) * 4;
    const int c = (int)((i4 / PLANE) % C_CH);   // PLANE % 4 == 0 -> uniform channel per float4
    const float mean = ws[2048 + c];
    const float istd = ws[2048 + C_CH + c];
    const float w    = wgt[c];
    const float b    = bias[c];

    // Last use of x + 205MB streaming store: keep both out of L2 (NT hints).
    v4f v = __builtin_nontemporal_load(reinterpret_cast<const v4f*>(x + i4));
    float r[4] = { v.x, v.y, v.z, v.w };
    #pragma unroll
    for (int j = 0; j < 4; ++j) {
        const uint32_t i = (uint32_t)i4 + (uint32_t)j;
        float xc = r[j] - mean;
        xc = fi(xc, i, 1u);
        float t = xc * istd;
        t = fi(t, i, 2u);
        t = t * w;
        t = fi(t, i, 3u);
        t = t + b;
        t = fi(t, i, 4u);
        r[j] = t;
    }
    v4f o = { r[0], r[1], r[2], r[3] };
    __builtin_nontemporal_store(o, reinterpret_cast<v4f*>(out + i4));
}

extern "C" void kernel_launch(void* const* d_in, const int* in_sizes, int n_in,
                              void* d_out, int out_size, void* d_ws, size_t ws_size,
                              hipStream_t stream) {
    const float* x    = (const float*)d_in[0];
    const float* wgt  = (const float*)d_in[1];
    const float* bias = (const float*)d_in[2];
    float* out = (float*)d_out;
    float* ws  = (float*)d_ws;

    dim3 gridR(C_CH, SPLIT);                       // 1024 blocks x 8 waves
    bn_reduce_kernel<<<gridR, 256, 0, stream>>>(x, ws);
    bn_stats_kernel<<<1, C_CH, 0, stream>>>(ws);
    const int nblk = (int)(TOTAL_ELEMS / 1024);    // 50176, exact
    bn_norm_kernel<<<nblk, 256, 0, stream>>>(x, wgt, bias, ws, out);
}